// LightGCN_81707457839463
// MI455X (gfx1250) — compile-verified
//
#include <hip/hip_runtime.h>
#include <hip/hip_bf16.h>

#define NU 100000
#define NI 50000
#define NN 150000           // NU + NI
#define DIM 64
#define NE 1000000
#define NLAYERS 3
#define WPB 8               // waves per block (256 threads / 32)

// ---------------------------------------------------------------- utilities
__global__ __launch_bounds__(1024) void k_zero_u32(unsigned* __restrict__ p, int n) {
    int i = blockIdx.x * blockDim.x + threadIdx.x;
    if (i < n) p[i] = 0u;
}

// x0 = concat(user_emb, item_emb); acc = x0   (float4 vectorized)
__global__ __launch_bounds__(256) void k_init(const float4* __restrict__ ue,
                                              const float4* __restrict__ ie,
                                              float4* __restrict__ x,
                                              float4* __restrict__ acc,
                                              int n4u, int n4tot) {
    int i = blockIdx.x * blockDim.x + threadIdx.x;
    if (i < n4tot) {
        float4 v = (i < n4u) ? ue[i] : ie[i - n4u];
        x[i] = v;
        acc[i] = v;
    }
}

// deg[n] = number of incident directed entries (both directions of each edge)
__global__ __launch_bounds__(256) void k_degree(const int* __restrict__ ei,
                                                unsigned* __restrict__ deg) {
    int e = blockIdx.x * blockDim.x + threadIdx.x;
    if (e >= NE) return;
    __builtin_prefetch(ei + e + 8192, 0, 3);            // global_prefetch on edge stream
    __builtin_prefetch(ei + NE + e + 8192, 0, 3);
    atomicAdd(&deg[ei[e]], 1u);
    atomicAdd(&deg[ei[NE + e]], 1u);
}

// d_inv = deg > 0 ? 1/sqrt(deg) : 0
__global__ __launch_bounds__(256) void k_dinv(const unsigned* __restrict__ deg,
                                              float* __restrict__ dinv, int n) {
    int i = blockIdx.x * blockDim.x + threadIdx.x;
    if (i >= n) return;
    unsigned d = deg[i];
    dinv[i] = d ? (1.0f / sqrtf((float)d)) : 0.0f;
}

// single-workgroup exclusive scan of deg -> offsets[0..n], duplicate into cursor
__global__ __launch_bounds__(1024) void k_scan(const unsigned* __restrict__ deg,
                                               unsigned* __restrict__ offs,
                                               unsigned* __restrict__ cur, int n) {
    __shared__ unsigned sm[1024];
    __shared__ unsigned srun;
    const int t = threadIdx.x;
    if (t == 0) srun = 0u;
    __syncthreads();
    for (int base = 0; base < n; base += 1024) {
        int i = base + t;
        unsigned v = (i < n) ? deg[i] : 0u;
        sm[t] = v;
        __syncthreads();
        #pragma unroll
        for (int off = 1; off < 1024; off <<= 1) {
            unsigned add = (t >= off) ? sm[t - off] : 0u;
            __syncthreads();
            sm[t] += add;
            __syncthreads();
        }
        unsigned run  = srun;
        unsigned excl = sm[t] - v;
        if (i < n) { offs[i] = run + excl; cur[i] = run + excl; }
        __syncthreads();
        if (t == 1023) srun = run + sm[1023];
        __syncthreads();
    }
    if (t == 0) { offs[n] = srun; cur[n] = srun; }
}

// scatter both directions of each edge into CSR col array via per-row cursors
__global__ __launch_bounds__(256) void k_scatter(const int* __restrict__ ei,
                                                 unsigned* __restrict__ cur,
                                                 int* __restrict__ cols) {
    int e = blockIdx.x * blockDim.x + threadIdx.x;
    if (e >= NE) return;
    __builtin_prefetch(ei + e + 8192, 0, 3);
    int u = ei[e];
    int v = ei[NE + e];
    unsigned pu = atomicAdd(&cur[u], 1u);
    cols[pu] = v;
    unsigned pv = atomicAdd(&cur[v], 1u);
    cols[pv] = u;
}

// ----------------------------------------------------------------- SpMM layer
// one wave per node; lane owns 2 of the 64 dims (float2 -> 256B coalesced row)
// y[r] = dinv[r] * sum_c dinv[c] * x[c];  acc += y;  final: acc = acc*0.25
__global__ __launch_bounds__(256) void k_spmm(const int* __restrict__ cols,
                                              const unsigned* __restrict__ offs,
                                              const float* __restrict__ dinv,
                                              const float* __restrict__ xin,
                                              float* __restrict__ xnext,
                                              float* __restrict__ acc,
                                              int final_layer) {
    __shared__ int scols[WPB][32];
    const int wave = threadIdx.x >> 5;
    const int lane = threadIdx.x & 31;
    const int node = blockIdx.x * WPB + wave;
    if (node >= NN) return;

    const unsigned start = offs[node];
    const unsigned end   = offs[node + 1];
    float2 sum; sum.x = 0.0f; sum.y = 0.0f;

    for (unsigned c0 = start; c0 < end; c0 += 32) {
        const int cnt = (int)((end - c0) < 32u ? (end - c0) : 32u);
#if defined(__gfx1250__)
        // CDNA5 async copy: stage this chunk of neighbor indices into LDS,
        // tracked with ASYNCcnt. (cols has +64 tail padding, so the lanes
        // beyond cnt over-read harmlessly.)
        {
            unsigned lds_off = (unsigned)(size_t)(&scols[wave][lane]);
            unsigned long long ga =
                (unsigned long long)(size_t)(cols + c0 + (unsigned)lane);
            asm volatile("global_load_async_to_lds_b32 %0, %1, off"
                         :: "v"(lds_off), "v"(ga) : "memory");
            asm volatile("s_wait_asynccnt 0x0" ::: "memory");
        }
#else
        scols[wave][lane] = cols[c0 + (unsigned)(lane < cnt ? lane : 0)];
        __threadfence_block();
#endif
        #pragma unroll 4
        for (int j = 0; j < cnt; ++j) {
            const int c = scols[wave][j];                      // LDS broadcast read
            const float s = dinv[c];                           // same-addr broadcast
            const float2* xr = (const float2*)(xin + ((size_t)c << 6));
            const float2 v = xr[lane];
            sum.x = fmaf(s, v.x, sum.x);
            sum.y = fmaf(s, v.y, sum.y);
        }
    }

    const float dr = dinv[node];
    float2 r; r.x = dr * sum.x; r.y = dr * sum.y;

    float2* accp = (float2*)(acc + ((size_t)node << 6));
    float2 a = accp[lane];
    a.x += r.x; a.y += r.y;
    if (final_layer) { a.x *= 0.25f; a.y *= 0.25f; }           // /(n_layers+1)
    accp[lane] = a;

    if (!final_layer) {
        float2* xo = (float2*)(xnext + ((size_t)node << 6));
        xo[lane] = r;
    }
}

// ------------------------------------------------------------------- launcher
extern "C" void kernel_launch(void* const* d_in, const int* in_sizes, int n_in,
                              void* d_out, int out_size, void* d_ws, size_t ws_size,
                              hipStream_t stream) {
    const float* ue = (const float*)d_in[0];   // [NU, 64]
    const float* ie = (const float*)d_in[1];   // [NI, 64]
    const int*   ei = (const int*)d_in[2];     // [2, NE]
    // d_in[3] = n_layers (device scalar, fixed to 3 by the reference setup)
    (void)in_sizes; (void)n_in; (void)out_size; (void)ws_size;

    char* p = (char*)d_ws;
    auto take = [&](size_t bytes) -> char* {
        char* r = p;
        p += (bytes + 255) & ~(size_t)255;
        return r;
    };
    unsigned* deg  = (unsigned*)take(sizeof(unsigned) * NN);
    unsigned* offs = (unsigned*)take(sizeof(unsigned) * (NN + 1));
    unsigned* cur  = (unsigned*)take(sizeof(unsigned) * (NN + 1));
    float*    dinv = (float*)take(sizeof(float) * NN);
    int*      cols = (int*)take(sizeof(int) * (2 * NE + 64));   // +tail pad for async over-read
    float*    xa   = (float*)take(sizeof(float) * (size_t)NN * DIM);
    float*    xb   = (float*)take(sizeof(float) * (size_t)NN * DIM);
    float*    acc  = (float*)d_out;

    k_zero_u32<<<(NN + 1023) / 1024, 1024, 0, stream>>>(deg, NN);
    k_init<<<(NN * (DIM / 4) + 255) / 256, 256, 0, stream>>>(
        (const float4*)ue, (const float4*)ie, (float4*)xa, (float4*)acc,
        NU * (DIM / 4), NN * (DIM / 4));
    k_degree<<<(NE + 255) / 256, 256, 0, stream>>>(ei, deg);
    k_dinv<<<(NN + 255) / 256, 256, 0, stream>>>(deg, dinv, NN);
    k_scan<<<1, 1024, 0, stream>>>(deg, offs, cur, NN);
    k_scatter<<<(NE + 255) / 256, 256, 0, stream>>>(ei, cur, cols);

    const float* xin = xa;
    float*       xnx = xb;
    for (int l = 0; l < NLAYERS; ++l) {
        int fin = (l == NLAYERS - 1);
        k_spmm<<<(NN + WPB - 1) / WPB, 256, 0, stream>>>(
            cols, offs, dinv, xin, xnx, acc, fin);
        const float* t = xnx; xnx = (float*)xin; xin = t;
    }
}